// GCNEmbedding_46617575030954
// MI455X (gfx1250) — compile-verified
//
#include <hip/hip_runtime.h>
#include <hip/hip_bf16.h>
#include <math.h>

// ---------------------------------------------------------------------------
// GCN 2-layer forward for MI455X (gfx1250, wave32).
// Dense transforms use V_WMMA_F32_16X16X4_F32 (native fp32 matrix op).
// Message passing uses edge-parallel float4 gathers + global_atomic_add_f32.
// ---------------------------------------------------------------------------

typedef float v2f __attribute__((ext_vector_type(2)));
typedef float v8f __attribute__((ext_vector_type(8)));

#define BN_EPS 1e-5f

// ---------------- degree / normalization ----------------

__global__ void k_init_deg(float* deg, int n) {
    int t = blockIdx.x * blockDim.x + threadIdx.x;
    if (t < n) deg[t] = 1.0f;   // self-loop weight
}

__global__ void k_accum_deg(const int* __restrict__ dst,
                            const float* __restrict__ w,
                            float* __restrict__ deg, int nE) {
    int t = blockIdx.x * blockDim.x + threadIdx.x;
    if (t < nE) atomicAdd(&deg[dst[t]], w[t]);
}

__global__ void k_finish_dinv(float* deg, int n) {
    int t = blockIdx.x * blockDim.x + threadIdx.x;
    if (t < n) {
        float d = deg[t];
        deg[t] = (d > 0.0f) ? rsqrtf(d) : 0.0f;
    }
}

// ---------------- WMMA fp32 GEMM: C[M,N] = A[M,K] @ B[K,N] ----------------
// One wave per 16x16 output tile. blockDim.x == 32.
// A-frag: lane l -> row m0 + (l&15); VGPR v -> K = k0 + v + 2*(l>>4)
// B-frag: lane l -> col n0 + (l&15); VGPR v -> K = k0 + v + 2*(l>>4)
// C/D   : lane l -> col n0 + (l&15); VGPR r -> row m0 + r + 8*(l>>4)

__global__ __launch_bounds__(32)
void k_gemm_wmma_f32(const float* __restrict__ A,
                     const float* __restrict__ B,
                     float* __restrict__ C,
                     int M, int K, int N) {
    const int lane = threadIdx.x;
    const int m0 = blockIdx.x * 16;
    const int n0 = blockIdx.y * 16;
    const int hi = lane >> 4;               // 0 or 1
    const int lo = lane & 15;
    int row = m0 + lo;
    if (row >= M) row = M - 1;              // clamp reads; keep EXEC all-ones
    const int col = n0 + lo;
    const int kh  = 2 * hi;

    v8f acc = {0.f, 0.f, 0.f, 0.f, 0.f, 0.f, 0.f, 0.f};

    const float* ap = A + (size_t)row * K + kh;
    const float* bp = B + (size_t)kh * N + col;
    const size_t bstep = (size_t)4 * N;

    for (int k0 = 0; k0 < K; k0 += 4) {
        v2f a, b;
        a.x = ap[0];
        a.y = ap[1];
        b.x = bp[0];
        b.y = bp[N];
        acc = __builtin_amdgcn_wmma_f32_16x16x4_f32(
                  false, a, false, b, (short)0, acc, false, false);
        ap += 4;
        bp += bstep;
    }

    const int mbase = m0 + 8 * hi;
#pragma unroll
    for (int r = 0; r < 8; ++r) {
        int m = mbase + r;
        if (m < M) C[(size_t)m * N + (size_t)col] = acc[r];
    }
}

// ---------------- aggregate init: out = bias + dinv^2 * h (self loop) -------

__global__ void k_init_agg(const float* __restrict__ hd,
                           const float* __restrict__ dinv,
                           const float* __restrict__ bias,
                           float* __restrict__ out,
                           int n, int C) {
    int t = blockIdx.x * blockDim.x + threadIdx.x;
    if (t >= n * C) return;
    int node = t / C;
    int c = t - node * C;
    float di = dinv[node];
    out[t] = bias[c] + di * di * hd[t];
}

// ---------------- edge scatter: out[dst] += norm * h[src] ------------------
// C/4 threads per edge, float4 per thread.

__global__ void k_scatter(const float* __restrict__ h,
                          const float* __restrict__ dinv,
                          const int* __restrict__ src,
                          const int* __restrict__ dst,
                          const float* __restrict__ w,
                          float* __restrict__ out,
                          int nE, int C) {
    int t = blockIdx.x * blockDim.x + threadIdx.x;
    int tpe = C >> 2;
    int e = t / tpe;
    if (e >= nE) return;
    int cg = (t - e * tpe) << 2;
    int s = src[e], d = dst[e];
    float nrm = dinv[s] * w[e] * dinv[d];
    const float4 hv = *reinterpret_cast<const float4*>(h + (size_t)s * C + cg);
    float* o = out + (size_t)d * C + cg;
    atomicAdd(o + 0, nrm * hv.x);
    atomicAdd(o + 1, nrm * hv.y);
    atomicAdd(o + 2, nrm * hv.z);
    atomicAdd(o + 3, nrm * hv.w);
}

// ---------------- BatchNorm statistics (one block per channel) -------------

__global__ __launch_bounds__(256)
void k_bn_stats(const float* __restrict__ h, float* __restrict__ stats,
                int n, int C) {
    const int c = blockIdx.x;
    float s = 0.f, s2 = 0.f;
    for (int i = threadIdx.x; i < n; i += blockDim.x) {
        float v = h[(size_t)i * C + c];
        s += v;
        s2 += v * v;
    }
    __shared__ float sh[256];
    __shared__ float sh2[256];
    sh[threadIdx.x] = s;
    sh2[threadIdx.x] = s2;
    __syncthreads();
    for (int ofs = 128; ofs > 0; ofs >>= 1) {
        if (threadIdx.x < ofs) {
            sh[threadIdx.x] += sh[threadIdx.x + ofs];
            sh2[threadIdx.x] += sh2[threadIdx.x + ofs];
        }
        __syncthreads();
    }
    if (threadIdx.x == 0) {
        float inv_n = 1.0f / (float)n;
        float mean = sh[0] * inv_n;
        float var = sh2[0] * inv_n - mean * mean;   // biased
        stats[c] = mean;
        stats[C + c] = var;
    }
}

// ---------------- BN apply + ReLU ------------------------------------------

__global__ void k_bn_relu(const float* __restrict__ hin,
                          const float* __restrict__ stats,
                          const float* __restrict__ gamma,
                          const float* __restrict__ beta,
                          float* __restrict__ hout,
                          int n, int C) {
    int t = blockIdx.x * blockDim.x + threadIdx.x;
    if (t >= n * C) return;
    int c = t % C;
    float mean = stats[c];
    float var = stats[C + c];
    float v = (hin[t] - mean) * rsqrtf(var + BN_EPS) * gamma[c] + beta[c];
    hout[t] = v > 0.f ? v : 0.f;
}

// ---------------------------------------------------------------------------

extern "C" void kernel_launch(void* const* d_in, const int* in_sizes, int n_in,
                              void* d_out, int out_size, void* d_ws, size_t ws_size,
                              hipStream_t stream) {
    const float* x        = (const float*)d_in[0];
    const int*   esrc     = (const int*)d_in[1];
    const int*   edst     = (const int*)d_in[2];
    const float* ew       = (const float*)d_in[3];
    const float* W1       = (const float*)d_in[4];
    const float* b1       = (const float*)d_in[5];
    const float* gamma    = (const float*)d_in[6];
    const float* beta     = (const float*)d_in[7];
    const float* W2       = (const float*)d_in[8];
    const float* b2       = (const float*)d_in[9];
    float*       out      = (float*)d_out;

    const int HID  = in_sizes[5];                 // 64
    const int OUTC = in_sizes[9];                 // 128
    const int INC  = in_sizes[4] / HID;           // 128
    const int N    = in_sizes[0] / INC;           // 50000
    const int E    = in_sizes[1];                 // 1000000

    // workspace layout (floats)
    float* ws = (float*)d_ws;
    size_t off = 0;
    float* dinv = ws + off; off += (size_t)((N + 63) / 64) * 64;
    float* h1d  = ws + off; off += (size_t)N * HID;   // dense x@W1, reused as h2
    float* h1a  = ws + off; off += (size_t)N * HID;   // aggregated conv1 output
    float* g    = ws + off; off += (size_t)N * OUTC;  // dense h2@W2
    float* stats = ws + off; off += 2 * (size_t)HID;
    (void)ws_size;

    const int TB = 256;

    // 1) degrees -> dinv
    k_init_deg<<<(N + TB - 1) / TB, TB, 0, stream>>>(dinv, N);
    k_accum_deg<<<(E + TB - 1) / TB, TB, 0, stream>>>(edst, ew, dinv, E);
    k_finish_dinv<<<(N + TB - 1) / TB, TB, 0, stream>>>(dinv, N);

    // 2) h1d = x @ W1  (fp32 WMMA)
    {
        dim3 grid((N + 15) / 16, HID / 16);
        k_gemm_wmma_f32<<<grid, 32, 0, stream>>>(x, W1, h1d, N, INC, HID);
    }

    // 3) conv1 aggregate: init (bias + self-loop) then edge scatter
    k_init_agg<<<((size_t)N * HID + TB - 1) / TB, TB, 0, stream>>>(h1d, dinv, b1, h1a, N, HID);
    {
        long long totT = (long long)E * (HID / 4);
        k_scatter<<<(totT + TB - 1) / TB, TB, 0, stream>>>(h1d, dinv, esrc, edst, ew, h1a, E, HID);
    }

    // 4) BatchNorm stats + apply + ReLU  (h2 reuses h1d buffer)
    k_bn_stats<<<HID, 256, 0, stream>>>(h1a, stats, N, HID);
    k_bn_relu<<<((size_t)N * HID + TB - 1) / TB, TB, 0, stream>>>(h1a, stats, gamma, beta, h1d, N, HID);

    // 5) g = h2 @ W2  (fp32 WMMA)
    {
        dim3 grid((N + 15) / 16, OUTC / 16);
        k_gemm_wmma_f32<<<grid, 32, 0, stream>>>(h1d, W2, g, N, HID, OUTC);
    }

    // 6) conv2 aggregate directly into d_out
    k_init_agg<<<((size_t)N * OUTC + TB - 1) / TB, TB, 0, stream>>>(g, dinv, b2, out, N, OUTC);
    {
        long long totT = (long long)E * (OUTC / 4);
        k_scatter<<<(totT + TB - 1) / TB, TB, 0, stream>>>(g, dinv, esrc, edst, ew, out, E, OUTC);
    }
}